// TimeSeriesPredictor_11759620456923
// MI455X (gfx1250) — compile-verified
//
#include <hip/hip_runtime.h>
#include <cstddef>

// ---------------------------------------------------------------------------
// Model dims (compile-time)
// ---------------------------------------------------------------------------
#define BB     16
#define SS     1024
#define DD     512
#define INNERD 1024
#define NROWS  (BB * SS)
#define KC     64   // GEMM k-chunk staged through LDS

typedef __attribute__((ext_vector_type(2))) float v2f;
typedef __attribute__((ext_vector_type(8))) float v8f;

// fp32 WMMA: D(16x16,f32) = A(16x4,f32) * B(4x16,f32) + C
__device__ __forceinline__ v8f wmma4(v2f a, v2f b, v8f c) {
  return __builtin_amdgcn_wmma_f32_16x16x4_f32(false, a, false, b, (short)0, c,
                                               false, false);
}

__device__ __forceinline__ float sigf(float x)  { return 1.f / (1.f + __expf(-x)); }
__device__ __forceinline__ float siluf(float x) { return x * sigf(x); }
__device__ __forceinline__ float logsigf(float x) {
  return fminf(x, 0.f) - log1pf(__expf(-fabsf(x)));
}
__device__ __forceinline__ float geluf(float x) {
  float x3 = x * x * x;
  return 0.5f * x * (1.f + tanhf(0.7978845608f * (x + 0.044715f * x3)));
}

// ---------------------------------------------------------------------------
// Generic WMMA fp32 GEMM:  C[M x Nc] = A[M x K] * B + bias + resid
//   opB==0: B stored K x Nc row-major (ldb); B panel staged into LDS with
//           async global->LDS b128 copies (double buffered, ASYNCcnt).
//   opB==1: B stored Nc x K row-major (ldb)  (y = x @ W^T), contiguous loads.
// One wave computes one 16x16 C tile; 4 waves per block stacked in M share B.
// ---------------------------------------------------------------------------
__global__ __launch_bounds__(128) void gemm_wmma_f32(
    const float* __restrict__ A, const float* __restrict__ Bw,
    const float* __restrict__ bias, const float* __restrict__ resid,
    float* __restrict__ C, int M, int Nc, int K, int lda, int ldb, int ldc,
    int opB) {
  __shared__ __align__(16) float bpan[2][KC * 16];
  int tid = threadIdx.x;
  int w = tid >> 5, lane = tid & 31;
  int mr = lane & 15, ko = (lane >> 4) * 2;
  int n0 = blockIdx.x * 16;
  int m0 = (blockIdx.y * 4 + w) * 16;
  int m0c = m0 < M ? m0 : M - 16;  // never exceeds in practice; keeps loads safe

  const float* arow = A + (size_t)(m0c + mr) * lda + ko;
  v8f acc = {};

  if (opB == 0) {
    const int nch = K / KC;
    // prime chunk 0 into buffer 0 (coalesced async b128 fills: 64x16 panel)
    {
      const float* g0 = Bw + n0;
#pragma unroll
      for (int i = 0; i < 2; ++i) {
        int e4 = i * 128 + tid;            // 256 float4 groups
        int r = e4 >> 2, c4 = (e4 & 3) << 2;
        const float* g = g0 + (size_t)r * ldb + c4;
        unsigned lo = (unsigned)(size_t)&bpan[0][r * 16 + c4];
        asm volatile("global_load_async_to_lds_b128 %0, %1, off"
                     :: "v"(lo), "v"(g) : "memory");
      }
    }
    asm volatile("s_wait_asynccnt 0x0" ::: "memory");
    __syncthreads();

    for (int ch = 0; ch < nch; ++ch) {
      int cur = ch & 1;
      if (ch + 1 < nch) {  // issue next panel into the other buffer
        const float* g0 = Bw + (size_t)(ch + 1) * KC * ldb + n0;
#pragma unroll
        for (int i = 0; i < 2; ++i) {
          int e4 = i * 128 + tid;
          int r = e4 >> 2, c4 = (e4 & 3) << 2;
          const float* g = g0 + (size_t)r * ldb + c4;
          unsigned lo = (unsigned)(size_t)&bpan[cur ^ 1][r * 16 + c4];
          asm volatile("global_load_async_to_lds_b128 %0, %1, off"
                       :: "v"(lo), "v"(g) : "memory");
        }
      }
      const float* ar = arow + ch * KC;
      const float* bp = &bpan[cur][0];
#pragma unroll
      for (int kk = 0; kk < KC; kk += 4) {
        v2f a = *(const v2f*)(ar + kk);
        v2f b;
        b.x = bp[(kk + ko) * 16 + mr];        // conflict-free LDS broadcast
        b.y = bp[(kk + ko + 1) * 16 + mr];
        acc = wmma4(a, b, acc);
      }
      asm volatile("s_wait_asynccnt 0x0" ::: "memory");
      __syncthreads();
    }
  } else {
    const float* brow = Bw + (size_t)(n0 + mr) * ldb + ko;
    __builtin_prefetch(brow, 0, 3);  // -> global_prefetch_b8
    for (int k0 = 0; k0 < K; k0 += 4) {
      v2f a = *(const v2f*)(arow + k0);
      v2f b = *(const v2f*)(brow + k0);
      acc = wmma4(a, b, acc);
    }
  }

  if (m0 < M) {
    int n = n0 + mr;
    float bv = bias ? bias[n] : 0.f;
#pragma unroll
    for (int r = 0; r < 8; ++r) {
      int m = m0 + r + (lane >> 4) * 8;
      float val = acc[r] + bv;
      if (resid) val += resid[(size_t)m * ldc + n];
      C[(size_t)m * ldc + n] = val;
    }
  }
}

// ---------------------------------------------------------------------------
// Row LayerNorm (optional bias), block=256, one row per block
// ---------------------------------------------------------------------------
__global__ __launch_bounds__(256) void ln_rows(const float* __restrict__ X,
                                               const float* __restrict__ w,
                                               const float* __restrict__ b,
                                               float* __restrict__ Y, int C) {
  __shared__ float sh[256];
  long row = blockIdx.x;
  int tid = threadIdx.x;
  const float* x = X + row * (long)C;
  float s = 0.f, s2 = 0.f;
  for (int c = tid; c < C; c += 256) {
    float v = x[c];
    s += v;
    s2 += v * v;
  }
  sh[tid] = s;
  __syncthreads();
  for (int o = 128; o > 0; o >>= 1) {
    if (tid < o) sh[tid] += sh[tid + o];
    __syncthreads();
  }
  float mu = sh[0] / C;
  __syncthreads();
  sh[tid] = s2;
  __syncthreads();
  for (int o = 128; o > 0; o >>= 1) {
    if (tid < o) sh[tid] += sh[tid + o];
    __syncthreads();
  }
  float var = sh[0] / C - mu * mu;
  float rs = rsqrtf(var + 1e-5f);
  float* y = Y + row * (long)C;
  for (int c = tid; c < C; c += 256) {
    float v = (x[c] - mu) * rs * w[c];
    if (b) v += b[c];
    y[c] = v;
  }
}

// ---------------------------------------------------------------------------
// Depthwise causal conv1d (K=3) + SiLU
// ---------------------------------------------------------------------------
__global__ void conv_silu(const float* __restrict__ X,
                          const float* __restrict__ Wc,
                          const float* __restrict__ bc, float* __restrict__ Y,
                          int C, int ldx, int ldy, long total) {
  long idx = (long)blockIdx.x * blockDim.x + threadIdx.x;
  if (idx >= total) return;
  int c = (int)(idx % C);
  long row = idx / C;
  int s = (int)(row % SS);
  const float* xp = X + row * (long)ldx + c;
  float acc = bc[c] + Wc[c * 3 + 2] * xp[0];
  if (s >= 1) acc += Wc[c * 3 + 1] * xp[-(long)ldx];
  if (s >= 2) acc += Wc[c * 3 + 0] * xp[-2 * (long)ldx];
  Y[row * (long)ldy + c] = siluf(acc);
}

// ---------------------------------------------------------------------------
// Block-diagonal 2x2 "headwise" projection (q/k/v of mLSTM)
// ---------------------------------------------------------------------------
__global__ void headwise2(const float* __restrict__ X,
                          const float* __restrict__ Wb, float* __restrict__ Y,
                          int ldx, long total) {
  long idx = (long)blockIdx.x * blockDim.x + threadIdx.x;
  if (idx >= total) return;
  int c = (int)(idx & 1023);
  long row = idx >> 10;
  int blk = c >> 1, o = c & 1;
  const float* x = X + row * (long)ldx + (blk << 1);
  const float* wp = Wb + blk * 4 + o * 2;
  Y[row * 1024 + c] = wp[0] * x[0] + wp[1] * x[1];
}

// ---------------------------------------------------------------------------
// ig/fg gate dot products over concat(q,k,v) (3072 dims); one wave per row
// ---------------------------------------------------------------------------
__global__ __launch_bounds__(256) void mlstm_gates(
    const float* __restrict__ Q, const float* __restrict__ K,
    const float* __restrict__ V, const float* __restrict__ igW,
    const float* __restrict__ igB, const float* __restrict__ fgW,
    const float* __restrict__ fgB, float* __restrict__ IG,
    float* __restrict__ FG, int Nr) {
  int w = threadIdx.x >> 5, lane = threadIdx.x & 31;
  int row = blockIdx.x * 8 + w;
  if (row >= Nr) return;
  const float* q = Q + (size_t)row * 1024;
  const float* k = K + (size_t)row * 1024;
  const float* v = V + (size_t)row * 1024;
  float ai = 0.f, af = 0.f;
  for (int c = lane; c < 1024; c += 32) {
    float qv = q[c], kv = k[c], vv = v[c];
    ai += qv * igW[c] + kv * igW[1024 + c] + vv * igW[2048 + c];
    af += qv * fgW[c] + kv * fgW[1024 + c] + vv * fgW[2048 + c];
  }
  for (int off = 16; off > 0; off >>= 1) {
    ai += __shfl_xor(ai, off);
    af += __shfl_xor(af, off);
  }
  if (lane == 0) {
    IG[row] = ai + igB[0];
    FG[row] = af + fgB[0];
  }
}

// ---------------------------------------------------------------------------
// Per-batch serial scan: lf = cumsum(logsigmoid(fg)),
// mmax_i = lf_i + runmax_j(ig_j - lf_j)   (analytic row-max of decay matrix)
// ---------------------------------------------------------------------------
__global__ void mlstm_decay_scan(const float* __restrict__ FG,
                                 const float* __restrict__ IG,
                                 float* __restrict__ LF,
                                 float* __restrict__ MM) {
  int b = threadIdx.x;
  if (b >= BB) return;
  float lfa = 0.f, mr = -1e30f;
  for (int t = 0; t < SS; ++t) {
    long idx = (long)b * SS + t;
    lfa += logsigf(FG[idx]);
    LF[idx] = lfa;
    mr = fmaxf(mr, IG[idx] - lfa);
    MM[idx] = lfa + mr;
  }
}

// ---------------------------------------------------------------------------
// mLSTM parallel matrix memory (flash-style, causal, precomputed row max).
// Block = 256 thr (8 waves) handles one 16-row q tile; each wave covers a
// 128-wide slice of the DH=1024 contraction (QK^T) and of the output (S@V).
// ---------------------------------------------------------------------------
__global__ __launch_bounds__(256) void mlstm_attn(
    const float* __restrict__ Q, const float* __restrict__ Km,
    const float* __restrict__ Vm, const float* __restrict__ LF,
    const float* __restrict__ IG, const float* __restrict__ MM,
    float* __restrict__ O) {
  int b = blockIdx.y;
  int i0 = blockIdx.x;
  int tid = threadIdx.x, w = tid >> 5, lane = tid & 31;
  int mr = lane & 15, ko = (lane >> 4) * 2;

  const float* Qb = Q + (size_t)b * SS * INNERD;
  const float* Kb = Km + (size_t)b * SS * INNERD;
  const float* Vb = Vm + (size_t)b * SS * INNERD;

  // q fragments for this wave's 128-wide K slice (64 VGPRs/lane)
  v2f qf[32];
  const float* qrow = Qb + (size_t)(i0 * 16 + mr) * INNERD + w * 128 + ko;
#pragma unroll
  for (int kk = 0; kk < 32; ++kk) qf[kk] = *(const v2f*)(qrow + kk * 4);

  v8f outacc[8];
  {
    v8f z = {};
#pragma unroll
    for (int st = 0; st < 8; ++st) outacc[st] = z;
  }

  __shared__ float scp[8][16][16];
  __shared__ float sfin[16][16];
  __shared__ float rowsum[16];
  __shared__ float lfi[16];
  if (tid < 16) {
    rowsum[tid] = 0.f;
    lfi[tid] = LF[(size_t)b * SS + i0 * 16 + tid];
  }
  __syncthreads();

  for (int j0 = 0; j0 <= i0; ++j0) {
    // --- partial scores over this wave's K slice ---
    v8f sacc = {};
    const float* krow = Kb + (size_t)(j0 * 16 + mr) * INNERD + w * 128 + ko;
#pragma unroll
    for (int kk = 0; kk < 32; ++kk) {
      v2f bf = *(const v2f*)(krow + kk * 4);
      sacc = wmma4(qf[kk], bf, sacc);
    }
#pragma unroll
    for (int r = 0; r < 8; ++r)
      scp[w][r + (lane >> 4) * 8][lane & 15] = sacc[r];
    __syncthreads();

    // --- reduce across waves, apply decay / causal mask ---
    {
      int m = tid >> 4, n = tid & 15;
      float s = 0.f;
#pragma unroll
      for (int ww = 0; ww < 8; ++ww) s += scp[ww][m][n];
      s *= 0.03125f;  // dh^-0.5, dh = 1024
      long jg = (long)b * SS + j0 * 16 + n;
      float d = __expf(lfi[m] - LF[jg] + IG[jg] -
                       MM[(size_t)b * SS + i0 * 16 + m]);
      if (j0 == i0 && n > m) d = 0.f;
      sfin[m][n] = s * d;
    }
    __syncthreads();

    if (tid < 16) {
      float rs = 0.f;
#pragma unroll
      for (int n = 0; n < 16; ++n) rs += sfin[tid][n];
      rowsum[tid] += rs;
    }

    // --- out += scores @ V (this wave's 128 output cols) ---
#pragma unroll
    for (int st = 0; st < 8; ++st) {
      int nc = w * 128 + st * 16 + (lane & 15);
      const float* vcol = Vb + (size_t)(j0 * 16) * INNERD + nc;
#pragma unroll
      for (int kk = 0; kk < 4; ++kk) {
        v2f af;
        af.x = sfin[mr][kk * 4 + ko];
        af.y = sfin[mr][kk * 4 + ko + 1];
        v2f bfv;
        bfv.x = vcol[(size_t)(kk * 4 + ko) * INNERD];
        bfv.y = vcol[(size_t)(kk * 4 + ko + 1) * INNERD];
        outacc[st] = wmma4(af, bfv, outacc[st]);
      }
    }
    // next iteration's scp/sfin writes happen only after barriers all waves
    // reach post-AV in program order -> no extra barrier needed here
  }
  __syncthreads();

  float* Ob = O + (size_t)b * SS * INNERD;
#pragma unroll
  for (int st = 0; st < 8; ++st) {
#pragma unroll
    for (int r = 0; r < 8; ++r) {
      int m = r + (lane >> 4) * 8;
      float mmi = MM[(size_t)b * SS + i0 * 16 + m];
      float nor = fmaxf(fabsf(rowsum[m]), __expf(-mmi)) + 1e-6f;
      Ob[(size_t)(i0 * 16 + m) * INNERD + w * 128 + st * 16 + (lane & 15)] =
          outacc[st][r] / nor;
    }
  }
}

// ---------------------------------------------------------------------------
// mLSTM epilogue: headnorm(1024) + skip*xca, then * SiLU(z); in-place on HT
// ---------------------------------------------------------------------------
__global__ __launch_bounds__(256) void hnorm_skip_gate(
    float* __restrict__ HT, const float* __restrict__ XCA,
    const float* __restrict__ XZ, const float* __restrict__ onw,
    const float* __restrict__ skip) {
  __shared__ float sh[256];
  long row = blockIdx.x;
  int tid = threadIdx.x;
  float* ht = HT + row * 1024;
  float s = 0.f, s2 = 0.f;
  for (int c = tid; c < 1024; c += 256) {
    float v = ht[c];
    s += v;
    s2 += v * v;
  }
  sh[tid] = s;
  __syncthreads();
  for (int o = 128; o > 0; o >>= 1) {
    if (tid < o) sh[tid] += sh[tid + o];
    __syncthreads();
  }
  float mu = sh[0] / 1024.f;
  __syncthreads();
  sh[tid] = s2;
  __syncthreads();
  for (int o = 128; o > 0; o >>= 1) {
    if (tid < o) sh[tid] += sh[tid + o];
    __syncthreads();
  }
  float var = sh[0] / 1024.f - mu * mu;
  float rs = rsqrtf(var + 1e-5f);
  const float* xca = XCA + row * 1024;
  const float* z = XZ + row * 2048 + 1024;
  for (int c = tid; c < 1024; c += 256) {
    float hn = (ht[c] - mu) * rs * onw[c];
    ht[c] = (hn + skip[c] * xca[c]) * siluf(z[c]);
  }
}

// ---------------------------------------------------------------------------
// GEGLU: g = gelu(u[:768]) * u[768:]
// ---------------------------------------------------------------------------
__global__ void geglu(const float* __restrict__ U, float* __restrict__ Y,
                      long total) {
  long idx = (long)blockIdx.x * blockDim.x + threadIdx.x;
  if (idx >= total) return;
  int c = (int)(idx % 768);
  long row = idx / 768;
  const float* u = U + row * 1536;
  Y[row * 768 + c] = geluf(u[c]) * u[768 + c];
}

// ---------------------------------------------------------------------------
// sLSTM recurrence: block per (head, batch); 4x128x128 recurrent weights in
// LDS (256KB of the 320KB WGP LDS); h_prev shared in LDS; 1024 serial steps.
// ---------------------------------------------------------------------------
__global__ __launch_bounds__(128) void slstm_scan(
    const float* __restrict__ GI, const float* __restrict__ GF,
    const float* __restrict__ GZ, const float* __restrict__ GO,
    const float* __restrict__ Ri, const float* __restrict__ Rf,
    const float* __restrict__ Rz, const float* __restrict__ Ro,
    const float* __restrict__ Bi, const float* __restrict__ Bf,
    const float* __restrict__ Bz, const float* __restrict__ Bo,
    float* __restrict__ YS) {
  int hh = blockIdx.x, b = blockIdx.y;
  int e = threadIdx.x;
  __shared__ float R[4][128][128];  // [gate][d][e]
  __shared__ float hsh[128];
  const float* Rg[4] = {Ri + (size_t)hh * 16384, Rf + (size_t)hh * 16384,
                        Rz + (size_t)hh * 16384, Ro + (size_t)hh * 16384};
  for (int g = 0; g < 4; ++g)
    for (int t = e; t < 16384; t += 128) R[g][t >> 7][t & 127] = Rg[g][t];
  hsh[e] = 0.f;
  __syncthreads();

  float bi = Bi[hh * 128 + e], bf = Bf[hh * 128 + e];
  float bz = Bz[hh * 128 + e], bo = Bo[hh * 128 + e];
  float cst = 0.f, nst = 0.f, mst = 0.f;

  for (int t = 0; t < SS; ++t) {
    size_t row = ((size_t)b * SS + t) * 512 + hh * 128 + e;
    float ai = GI[row] + bi, af = GF[row] + bf;
    float az = GZ[row] + bz, ao = GO[row] + bo;
#pragma unroll 8
    for (int d = 0; d < 128; ++d) {
      float hv = hsh[d];
      ai += hv * R[0][d][e];
      af += hv * R[1][d][e];
      az += hv * R[2][d][e];
      ao += hv * R[3][d][e];
    }
    float logf = mst + logsigf(af);
    float mn = fmaxf(ai, logf);
    float iex = __expf(ai - mn), fex = __expf(logf - mn);
    cst = fex * cst + iex * tanhf(az);
    nst = fex * nst + iex;
    mst = mn;
    float hnew = sigf(ao) * cst / nst;
    __syncthreads();
    hsh[e] = hnew;
    YS[row] = hnew;
    __syncthreads();
  }
}

// ---------------------------------------------------------------------------
// Per-head groupnorm(128) * gn_w, residual-added into H
// ---------------------------------------------------------------------------
__global__ __launch_bounds__(128) void groupnorm_resid(
    const float* __restrict__ YS, const float* __restrict__ gnw,
    float* __restrict__ H) {
  __shared__ float sh[128];
  long row = blockIdx.x;
  int hh = blockIdx.y, e = threadIdx.x;
  float v = YS[row * 512 + hh * 128 + e];
  sh[e] = v;
  __syncthreads();
  for (int o = 64; o > 0; o >>= 1) {
    if (e < o) sh[e] += sh[e + o];
    __syncthreads();
  }
  float mu = sh[0] / 128.f;
  __syncthreads();
  sh[e] = v * v;
  __syncthreads();
  for (int o = 64; o > 0; o >>= 1) {
    if (e < o) sh[e] += sh[e + o];
    __syncthreads();
  }
  float var = sh[0] / 128.f - mu * mu;
  float rs = rsqrtf(var + 1e-5f);
  H[row * 512 + hh * 128 + e] += (v - mu) * rs * gnw[hh * 128 + e];
}

// ---------------------------------------------------------------------------
// Final head: out[b, :24] = h[b, S-1, :] @ out_W + out_b
// ---------------------------------------------------------------------------
__global__ void out_head(const float* __restrict__ H,
                         const float* __restrict__ Wo,
                         const float* __restrict__ bo, float* __restrict__ out) {
  int b = blockIdx.x, j = threadIdx.x;
  if (j >= 24) return;
  const float* hr = H + ((size_t)b * SS + SS - 1) * 512;
  float acc = bo[j];
  for (int c = 0; c < 512; ++c) acc += hr[c] * Wo[c * 24 + j];
  out[b * 24 + j] = acc;
}

// ---------------------------------------------------------------------------
// Host orchestration
// ---------------------------------------------------------------------------
extern "C" void kernel_launch(void* const* d_in, const int* in_sizes, int n_in,
                              void* d_out, int out_size, void* d_ws,
                              size_t ws_size, hipStream_t stream) {
  (void)in_sizes; (void)n_in; (void)out_size; (void)ws_size;
  auto in = [&](int i) { return (const float*)d_in[i]; };

  const int Nr = NROWS;
  float* W = (float*)d_ws;
  size_t off = 0;
  auto alloc = [&](size_t n) { float* p = W + off; off += n; return p; };
  float* h   = alloc((size_t)Nr * 512);
  float* tA  = alloc((size_t)Nr * 512);
  float* xz  = alloc((size_t)Nr * 2048);
  float* xca = alloc((size_t)Nr * 1024);
  float* qb  = alloc((size_t)Nr * 1024);
  float* kb  = alloc((size_t)Nr * 1024);
  float* vb  = alloc((size_t)Nr * 1024);
  float* ht  = alloc((size_t)Nr * 1024);
  float* ig  = alloc(Nr);
  float* fg  = alloc(Nr);
  float* lfb = alloc(Nr);
  float* mmb = alloc(Nr);

  auto G = [&](const float* A, const float* Bw, const float* bias,
               const float* resid, float* C, int M, int Nc, int K, int lda,
               int ldb, int ldc, int opB) {
    dim3 grid(Nc / 16, M / 64);
    gemm_wmma_f32<<<grid, 128, 0, stream>>>(A, Bw, bias, resid, C, M, Nc, K,
                                            lda, ldb, ldc, opB);
  };

  // embedding: h = x @ emb_W + emb_b
  G(in(0), in(1), in(2), nullptr, h, Nr, 512, 64, 64, 512, 512, 0);

  auto mlstm = [&](int pi) {
    const float *lnw = in(pi), *upW = in(pi + 1), *cw = in(pi + 2),
                *cb = in(pi + 3), *qW = in(pi + 4), *kW = in(pi + 5),
                *vW = in(pi + 6), *igW = in(pi + 7), *igB = in(pi + 8),
                *fgW = in(pi + 9), *fgB = in(pi + 10), *skip = in(pi + 11),
                *onw = in(pi + 12), *dW = in(pi + 13);
    ln_rows<<<Nr, 256, 0, stream>>>(h, lnw, nullptr, tA, 512);
    G(tA, upW, nullptr, nullptr, xz, Nr, 2048, 512, 512, 2048, 2048, 0);
    long tot = (long)Nr * 1024;
    conv_silu<<<(int)((tot + 255) / 256), 256, 0, stream>>>(xz, cw, cb, xca,
                                                            1024, 2048, 1024,
                                                            tot);
    headwise2<<<(int)((tot + 255) / 256), 256, 0, stream>>>(xca, qW, qb, 1024,
                                                            tot);
    headwise2<<<(int)((tot + 255) / 256), 256, 0, stream>>>(xca, kW, kb, 1024,
                                                            tot);
    headwise2<<<(int)((tot + 255) / 256), 256, 0, stream>>>(xz, vW, vb, 2048,
                                                            tot);
    mlstm_gates<<<Nr / 8, 256, 0, stream>>>(qb, kb, vb, igW, igB, fgW, fgB, ig,
                                            fg, Nr);
    mlstm_decay_scan<<<1, BB, 0, stream>>>(fg, ig, lfb, mmb);
    mlstm_attn<<<dim3(SS / 16, BB), 256, 0, stream>>>(qb, kb, vb, lfb, ig, mmb,
                                                      ht);
    hnorm_skip_gate<<<Nr, 256, 0, stream>>>(ht, xca, xz, onw, skip);
    G(ht, dW, nullptr, h, h, Nr, 512, 1024, 1024, 512, 512, 0);
  };

  mlstm(3);  // block 0 (m0)

  {  // block 1: sLSTM
    const float *lnw = in(17), *cw = in(18), *cb = in(19);
    const float *wi = in(20), *wf = in(21), *wz = in(22), *wo = in(23);
    const float *ri = in(24), *rf = in(25), *rz = in(26), *ro = in(27);
    const float *bi = in(28), *bfp = in(29), *bz = in(30), *bo = in(31);
    const float* gnw = in(32);
    ln_rows<<<Nr, 256, 0, stream>>>(h, lnw, nullptr, tA, 512);
    long tot = (long)Nr * 512;
    conv_silu<<<(int)((tot + 255) / 256), 256, 0, stream>>>(tA, cw, cb, xca,
                                                            512, 512, 512, tot);
    for (int hh = 0; hh < 4; ++hh) {
      G(xca + hh * 128, wi + (size_t)hh * 16384, nullptr, nullptr,
        qb + hh * 128, Nr, 128, 128, 512, 128, 512, 1);
      G(xca + hh * 128, wf + (size_t)hh * 16384, nullptr, nullptr,
        kb + hh * 128, Nr, 128, 128, 512, 128, 512, 1);
      G(tA + hh * 128, wz + (size_t)hh * 16384, nullptr, nullptr,
        vb + hh * 128, Nr, 128, 128, 512, 128, 512, 1);
      G(tA + hh * 128, wo + (size_t)hh * 16384, nullptr, nullptr,
        ht + hh * 128, Nr, 128, 128, 512, 128, 512, 1);
    }
    slstm_scan<<<dim3(4, BB), 128, 0, stream>>>(qb, kb, vb, ht, ri, rf, rz, ro,
                                                bi, bfp, bz, bo, xz);
    groupnorm_resid<<<dim3(Nr, 4), 128, 0, stream>>>(xz, gnw, h);
  }

  {  // block 1: gated FFN
    ln_rows<<<Nr, 256, 0, stream>>>(h, in(33), nullptr, tA, 512);
    G(tA, in(34), nullptr, nullptr, xz, Nr, 1536, 512, 512, 1536, 1536, 0);
    long tot = (long)Nr * 768;
    geglu<<<(int)((tot + 255) / 256), 256, 0, stream>>>(xz, xca, tot);
    G(xca, in(35), nullptr, h, h, Nr, 512, 768, 768, 512, 512, 0);
  }

  mlstm(36);  // block 2 (m2)

  // post norms + output head
  ln_rows<<<Nr, 256, 0, stream>>>(h, in(50), nullptr, tA, 512);
  ln_rows<<<Nr, 256, 0, stream>>>(tA, in(51), in(52), h, 512);
  out_head<<<BB, 32, 0, stream>>>(h, in(53), in(54), (float*)d_out);
}